// RNNAE_41755672052142
// MI455X (gfx1250) — compile-verified
//
#include <hip/hip_runtime.h>
#include <math.h>

// ---------------------------------------------------------------------------
// RNNAE rollout: 5 tasks x 10 steps of an H=10 LSTM + 3-layer MLP, f32.
// Latency-bound micro-problem -> single persistent wave32 workgroup.
// All GEMMs run on the CDNA5 matrix pipe via V_WMMA_F32_16X16X4_F32
// (full f32, matching the reference precision). Weights staged in LDS once.
// ---------------------------------------------------------------------------

typedef __attribute__((ext_vector_type(2))) float v2f;
typedef __attribute__((ext_vector_type(8))) float v8f;

#define HID 10
#define NT  5
#define NS  10

// ---- WMMA f32 16x16x4 fragment helpers (layouts per ISA 7.12.2) -----------
// A (16x4, MxK): lane&15 = row M; lanes>=16 hold K+2; v.x=K, v.y=K+1
__device__ __forceinline__ v2f ldsA(const float* m, int ld, int koff) {
  int lane = threadIdx.x;
  int row = lane & 15;
  int kb  = koff + ((lane >> 4) << 1);
  v2f a; a.x = m[row * ld + kb]; a.y = m[row * ld + kb + 1];
  return a;
}
// B (4x16, KxN): lane&15 = col N; half-lane split on K mirrors A
__device__ __forceinline__ v2f ldsB(const float* m, int ld, int koff, int noff) {
  int lane = threadIdx.x;
  int col = noff + (lane & 15);
  int kb  = koff + ((lane >> 4) << 1);
  v2f b; b.x = m[kb * ld + col]; b.y = m[(kb + 1) * ld + col];
  return b;
}
// D (16x16): VGPR r -> row (r + 8*(lane>=16)), col = lane&15
__device__ __forceinline__ void stD(float* m, int ld, int noff, v8f d) {
  int lane = threadIdx.x;
  int col = noff + (lane & 15);
  int rb  = (lane >> 4) << 3;
#pragma unroll
  for (int r = 0; r < 8; ++r) m[(rb + r) * ld + col] = d[r];
}

__device__ __forceinline__ v8f wmma4(v2f a, v2f b, v8f c) {
  // 8-arg pattern: (neg_a, A, neg_b, B, c_mod, C, reuse_a, reuse_b)
  return __builtin_amdgcn_wmma_f32_16x16x4_f32(false, a, false, b, (short)0, c,
                                               false, false);
}

__device__ __forceinline__ float sigmf(float x) { return 1.0f / (1.0f + expf(-x)); }

__global__ __launch_bounds__(32) void rnnae_kernel(
    const float* __restrict__ omega,   // [5]
    const float* __restrict__ noise,   // [5,10]
    const unsigned char* __restrict__ go1,  // [10] bool mask
    const unsigned char* __restrict__ go2,  // [10] bool mask
    const float* __restrict__ W_ih,    // [40,2]
    const float* __restrict__ W_hh,    // [40,10]
    const float* __restrict__ b_ih,    // [40]
    const float* __restrict__ b_hh,    // [40]
    const float* __restrict__ w1,      // [20,10]
    const float* __restrict__ b1,      // [20]
    const float* __restrict__ w2,      // [20,20]
    const float* __restrict__ b2,      // [20]
    const float* __restrict__ w3,      // [1,20]
    const float* __restrict__ b3,      // [1]
    float* __restrict__ out)           // [1]
{
  // ---- LDS staging (padded for 16x16x4 tiling; B-side padding is zeroed) --
  __shared__ float sWihT[4 * 48];    // (K=2 pad 4) x (N=40 pad 48)
  __shared__ float sWhhT[12 * 48];   // (K=10 pad 12) x 48
  __shared__ float sBias[48];        // b_ih + b_hh, pad 48
  __shared__ float sW1T[12 * 32];    // (K=10 pad 12) x (N=20 pad 32)
  __shared__ float sB1[32];
  __shared__ float sW2T[20 * 32];    // K=20 exact x pad 32
  __shared__ float sB2[32];
  __shared__ float sW3T[20 * 16];    // K=20 x (N=1 pad 16)
  __shared__ float sX[16 * 4];       // step input [s, ah, 0, 0]
  __shared__ float sH[16 * 12];      // hidden, K-padded
  __shared__ float sC[16 * 12];      // cell
  __shared__ float sG[16 * 48];      // gates
  __shared__ float sH1[16 * 32];
  __shared__ float sH2[16 * 32];
  __shared__ float sAh[NT * NS];
  __shared__ float sS[8], sErr[8], sOmega[8];

  const int lane = threadIdx.x;

  // ---- zero everything that acts as padding / running state --------------
  for (int i = lane; i < 4 * 48;  i += 32) sWihT[i] = 0.f;
  for (int i = lane; i < 12 * 48; i += 32) sWhhT[i] = 0.f;
  for (int i = lane; i < 48;      i += 32) sBias[i] = 0.f;
  for (int i = lane; i < 12 * 32; i += 32) sW1T[i] = 0.f;
  for (int i = lane; i < 32;      i += 32) { sB1[i] = 0.f; sB2[i] = 0.f; }
  for (int i = lane; i < 20 * 32; i += 32) sW2T[i] = 0.f;
  for (int i = lane; i < 20 * 16; i += 32) sW3T[i] = 0.f;
  for (int i = lane; i < 16 * 4;  i += 32) sX[i] = 0.f;
  for (int i = lane; i < 16 * 12; i += 32) { sH[i] = 0.f; sC[i] = 0.f; }
  if (lane < 8) { sS[lane] = 0.f; sErr[lane] = 0.f; sOmega[lane] = 0.f; }
  __syncthreads();

  // ---- load + transpose weights into LDS ---------------------------------
  for (int i = lane; i < 80;  i += 32) { int n = i >> 1, k = i & 1;  sWihT[k * 48 + n] = W_ih[i]; }
  for (int i = lane; i < 400; i += 32) { int n = i / 10, k = i % 10; sWhhT[k * 48 + n] = W_hh[i]; }
  for (int i = lane; i < 40;  i += 32) sBias[i] = b_ih[i] + b_hh[i];
  for (int i = lane; i < 200; i += 32) { int n = i / 10, k = i % 10; sW1T[k * 32 + n] = w1[i]; }
  for (int i = lane; i < 20;  i += 32) { sB1[i] = b1[i]; sB2[i] = b2[i]; }
  for (int i = lane; i < 400; i += 32) { int n = i / 20, k = i % 20; sW2T[k * 32 + n] = w2[i]; }
  for (int i = lane; i < 20;  i += 32) sW3T[i * 16] = w3[i];
  if (lane < NT) sOmega[lane] = omega[lane];

  // human() input precompute: ah[task, step]
  for (int i = lane; i < NT * NS; i += 32) {
    int r = i / NS, t = i % NS;
    float om = omega[r];
    bool g1 = (fabsf(om - 0.4f) < 1e-5f) || (fabsf(om - 0.8f) < 1e-5f);
    float scale = g1 ? 0.2f : 0.4f;
    bool msk = g1 ? (go1[t] != 0) : (go2[t] != 0);
    sAh[i] = (msk ? (om - 0.6f) / scale : 0.f) + noise[i];
  }
  const float b3v = b3[0];
  __syncthreads();

  // ---- sequential rollout -------------------------------------------------
  for (int t = 0; t < NS; ++t) {
    // x = [s, ah_t] (rows 5..15 and cols 2..3 stay zero)
    if (lane < 16) {
      sX[lane * 4 + 0] = (lane < NT) ? sS[lane] : 0.f;
      sX[lane * 4 + 1] = (lane < NT) ? sAh[lane * NS + t] : 0.f;
    }
    __syncthreads();

    // gates[5,40] = x@WihT + h@WhhT + bias  (3 N-tiles x 4 WMMAs)
#pragma unroll
    for (int nt = 0; nt < 3; ++nt) {
      float bv = sBias[nt * 16 + (lane & 15)];
      v8f acc;
#pragma unroll
      for (int r = 0; r < 8; ++r) acc[r] = bv;
      acc = wmma4(ldsA(sX, 4, 0), ldsB(sWihT, 48, 0, nt * 16), acc);
#pragma unroll
      for (int kc = 0; kc < 3; ++kc)
        acc = wmma4(ldsA(sH, 12, kc * 4), ldsB(sWhhT, 48, kc * 4, nt * 16), acc);
      stD(sG, 48, nt * 16, acc);
    }
    __syncthreads();

    // LSTM cell elementwise (gate order i,f,g,o)
    for (int e = lane; e < NT * HID; e += 32) {
      int r = e / HID, j = e % HID;
      float gi = sigmf(sG[r * 48 + j]);
      float gf = sigmf(sG[r * 48 + HID + j]);
      float gg = tanhf(sG[r * 48 + 2 * HID + j]);
      float go = sigmf(sG[r * 48 + 3 * HID + j]);
      float c  = gf * sC[r * 12 + j] + gi * gg;
      sC[r * 12 + j] = c;
      sH[r * 12 + j] = go * tanhf(c);
    }
    __syncthreads();

    // h1 = relu(h @ w1T + b1)
#pragma unroll
    for (int nt = 0; nt < 2; ++nt) {
      float bv = sB1[nt * 16 + (lane & 15)];
      v8f acc;
#pragma unroll
      for (int r = 0; r < 8; ++r) acc[r] = bv;
#pragma unroll
      for (int kc = 0; kc < 3; ++kc)
        acc = wmma4(ldsA(sH, 12, kc * 4), ldsB(sW1T, 32, kc * 4, nt * 16), acc);
      stD(sH1, 32, nt * 16, acc);
    }
    __syncthreads();
    for (int e = lane; e < NT * 2 * HID; e += 32) {
      int r = e / (2 * HID), j = e % (2 * HID);
      float v = sH1[r * 32 + j];
      sH1[r * 32 + j] = v > 0.f ? v : 0.f;
    }
    __syncthreads();

    // h2 = relu(h1 @ w2T + b2)
#pragma unroll
    for (int nt = 0; nt < 2; ++nt) {
      float bv = sB2[nt * 16 + (lane & 15)];
      v8f acc;
#pragma unroll
      for (int r = 0; r < 8; ++r) acc[r] = bv;
#pragma unroll
      for (int kc = 0; kc < 5; ++kc)
        acc = wmma4(ldsA(sH1, 32, kc * 4), ldsB(sW2T, 32, kc * 4, nt * 16), acc);
      stD(sH2, 32, nt * 16, acc);
    }
    __syncthreads();
    for (int e = lane; e < NT * 2 * HID; e += 32) {
      int r = e / (2 * HID), j = e % (2 * HID);
      float v = sH2[r * 32 + j];
      sH2[r * 32 + j] = v > 0.f ? v : 0.f;
    }
    __syncthreads();

    // ar = h2 @ w3T + b3  (single N-tile; result rows 0..4, col 0 -> lane 0)
    {
      v8f acc;
      float bi = ((lane & 15) == 0) ? b3v : 0.f;
#pragma unroll
      for (int r = 0; r < 8; ++r) acc[r] = bi;
#pragma unroll
      for (int kc = 0; kc < 5; ++kc)
        acc = wmma4(ldsA(sH2, 32, kc * 4), ldsB(sW3T, 16, kc * 4, 0), acc);
      if (lane == 0) {
#pragma unroll
        for (int r = 0; r < NT; ++r) {
          float s = sS[r] + acc[r];
          sS[r] = s;
          float d = s - sOmega[r];
          sErr[r] += d * d;
        }
      }
    }
    __syncthreads();
  }

  if (lane == 0) {
    float q = 0.f;
#pragma unroll
    for (int r = 0; r < NT; ++r) q += sErr[r];
    out[0] = q;
  }
}

extern "C" void kernel_launch(void* const* d_in, const int* in_sizes, int n_in,
                              void* d_out, int out_size, void* d_ws, size_t ws_size,
                              hipStream_t stream) {
  (void)in_sizes; (void)n_in; (void)out_size; (void)d_ws; (void)ws_size;
  const float* omega = (const float*)d_in[0];
  const float* noise = (const float*)d_in[1];
  const unsigned char* go1 = (const unsigned char*)d_in[2];  // jnp bool (1B)
  const unsigned char* go2 = (const unsigned char*)d_in[3];
  const float* W_ih = (const float*)d_in[4];
  const float* W_hh = (const float*)d_in[5];
  const float* b_ih = (const float*)d_in[6];
  const float* b_hh = (const float*)d_in[7];
  const float* w1 = (const float*)d_in[8];
  const float* b1 = (const float*)d_in[9];
  const float* w2 = (const float*)d_in[10];
  const float* b2 = (const float*)d_in[11];
  const float* w3 = (const float*)d_in[12];
  const float* b3 = (const float*)d_in[13];
  rnnae_kernel<<<1, 32, 0, stream>>>(omega, noise, go1, go2, W_ih, W_hh, b_ih,
                                     b_hh, w1, b1, w2, b2, w3, b3,
                                     (float*)d_out);
}